// SAGEConv_40364102647896
// MI455X (gfx1250) — compile-verified
//
#include <hip/hip_runtime.h>

// Problem constants (match reference file)
#define N_NODES   50000
#define N_EDGES   800000
#define D_NODE    64
#define D_ATTR    16
#define D_T       16
#define D_MSG     96          // D_NODE + D_ATTR + D_T
#define D_IN_ALL  160         // D_MSG + D_NODE (combined K)
#define D_OUT     128
#define LDS_PITCH 164         // 164 % 64 == 36 -> conflict-free 16-row column reads
#define ROWS_PB   32          // rows per block (2 M-tiles share every B fragment)

typedef __attribute__((ext_vector_type(2))) float v2f;
typedef __attribute__((ext_vector_type(4))) float v4f;
typedef __attribute__((ext_vector_type(8))) float v8f;

// ---------------------------------------------------------------------------
// Kernel 0: zero the aggregation buffer (d_ws). Must run every launch since
// the scatter kernel accumulates with atomics.
// ---------------------------------------------------------------------------
__global__ void zero_agg_kernel(v4f* __restrict__ agg4, int n4) {
    int i = blockIdx.x * blockDim.x + threadIdx.x;
    if (i < n4) agg4[i] = (v4f){0.f, 0.f, 0.f, 0.f};
}

// ---------------------------------------------------------------------------
// Kernel 1: edge scatter.  Thread t handles edge e = t/24, channel quad
// q = t%24 (channels 4q..4q+3 of the 96-wide message).  One 16B load +
// four global_atomic_add_f32.  Consecutive lanes touch consecutive 16B
// segments of the same destination row -> per-cacheline atomic merging,
// and agg (19.2 MB) is resident in the 192 MB L2.
// edge_attr / edge_t are touch-once streams -> non-temporal loads so they
// don't evict the hot x gather region or the atomic-resident agg from L2.
// ---------------------------------------------------------------------------
__global__ void edge_scatter_kernel(const float*     __restrict__ x,
                                    const long long* __restrict__ ei,   // [2, E] int64
                                    const float*     __restrict__ ea,   // [E, 16]
                                    const float*     __restrict__ et,   // [E, 16]
                                    float*           __restrict__ agg)  // [N, 96]
{
    long t = (long)blockIdx.x * blockDim.x + threadIdx.x;
    const long total = (long)N_EDGES * 24;
    if (t >= total) return;

    int e = (int)(t / 24);
    int q = (int)(t - (long)e * 24);   // 0..23
    int c = q * 4;                     // channel base 0..92

    long dst = (long)ei[e];            // row  (scatter destination)
    long src = (long)ei[N_EDGES + e];  // col  (gather source node)

    v4f v;
    if (c < D_NODE) {
        v = ((const v4f*)(x + src * D_NODE))[q];                       // hot, keep RT
    } else if (c < D_NODE + D_ATTR) {
        v = __builtin_nontemporal_load(
                (const v4f*)(ea + (long)e * D_ATTR) + (q - D_NODE / 4));
    } else {
        v = __builtin_nontemporal_load(
                (const v4f*)(et + (long)e * D_T) + (q - (D_NODE + D_ATTR) / 4));
    }

    float* dptr = agg + dst * D_MSG + c;
    atomicAdd(dptr + 0, v.x);
    atomicAdd(dptr + 1, v.y);
    atomicAdd(dptr + 2, v.z);
    atomicAdd(dptr + 3, v.w);
}

// ---------------------------------------------------------------------------
// Kernel 2: out = (0.5*agg) @ Wm + x @ Wr + (0.5*bm + br), via
// v_wmma_f32_16x16x4_f32.  One block = 32 output rows (two 16-row M-tiles);
// 8 waves each own a 16-column tile of the 128 output columns.  Each B
// fragment (from Wm/Wr, 80 KB total, L2/WGP$-resident) feeds TWO WMMAs,
// halving weight traffic vs one M-tile per block.  The 0.5 factor is folded
// into the A-staging; the bias sum seeds both C accumulators.
//
// Fragment layouts (ISA 7.12.2, f32):
//   A 16x4 : lane m = lane&15, VGPR{0,1} = A[m][kb + 2*(lane>>4) + {0,1}]
//   B 4x16 : lane n = lane&15, VGPR{0,1} = B[kb + 2*(lane>>4) + {0,1}][n]
//   C/D    : VGPR v -> M = v + 8*(lane>>4), N = lane&15
//
// WMMA always executes with EXEC all-ones: staging clamps tail rows, and
// only the epilogue stores are guarded.
// ---------------------------------------------------------------------------
__global__ void __launch_bounds__(256)
gemm_wmma_kernel(const float* __restrict__ agg,   // [N, 96]
                 const float* __restrict__ x,     // [N, 64]
                 const float* __restrict__ Wm,    // [96, 128]
                 const float* __restrict__ bm,    // [128]
                 const float* __restrict__ Wr,    // [64, 128]
                 const float* __restrict__ br,    // [128]
                 float*       __restrict__ out)   // [N, 128]
{
    __shared__ float ldsA[ROWS_PB * LDS_PITCH];

    const int m0 = blockIdx.x * ROWS_PB;

    // Stage combined A tile: cols 0..95 = 0.5*agg row, cols 96..159 = x row.
    // 5120 elements / 256 threads = 20 each (uniform trip count).
    for (int idx = threadIdx.x; idx < ROWS_PB * D_IN_ALL; idx += 256) {
        int r   = idx / D_IN_ALL;
        int cc  = idx - r * D_IN_ALL;
        int row = m0 + r;
        if (row >= N_NODES) row = N_NODES - 1;   // tail clamp (values unused)
        float v = (cc < D_MSG)
                ? 0.5f * agg[(long)row * D_MSG + cc]
                : x[(long)row * D_NODE + (cc - D_MSG)];
        ldsA[r * LDS_PITCH + cc] = v;
    }
    __syncthreads();

    const int lane  = threadIdx.x & 31;
    const int wave  = threadIdx.x >> 5;     // 0..7
    const int nlane = lane & 15;            // N (and A row M) within tile
    const int khalf = lane >> 4;            // 0 or 1 -> K sub-pair
    const int n     = wave * 16 + nlane;    // global output column

    const float bias = 0.5f * bm[n] + br[n];
    v8f c0 = {bias, bias, bias, bias, bias, bias, bias, bias};
    v8f c1 = c0;

    const float* arow0 = &ldsA[nlane * LDS_PITCH];
    const float* arow1 = arow0 + 16 * LDS_PITCH;

    // K = 96 over (0.5*agg) x Wm  -- one B fragment drives both M-tiles
    #pragma unroll
    for (int kb = 0; kb < D_MSG; kb += 4) {
        const int kk = kb + khalf * 2;
        v2f b;
        b.x = Wm[(kk + 0) * D_OUT + n];
        b.y = Wm[(kk + 1) * D_OUT + n];
        v2f a0 = *(const v2f*)(arow0 + kk);
        v2f a1 = *(const v2f*)(arow1 + kk);
        c0 = __builtin_amdgcn_wmma_f32_16x16x4_f32(false, a0, false, b,
                                                   (short)0, c0, false, false);
        c1 = __builtin_amdgcn_wmma_f32_16x16x4_f32(false, a1, false, b,
                                                   (short)0, c1, false, false);
    }
    // K = 64 over x x Wr
    #pragma unroll
    for (int kb = 0; kb < D_NODE; kb += 4) {
        const int kk = kb + khalf * 2;
        v2f b;
        b.x = Wr[(kk + 0) * D_OUT + n];
        b.y = Wr[(kk + 1) * D_OUT + n];
        v2f a0 = *(const v2f*)(arow0 + D_MSG + kk);
        v2f a1 = *(const v2f*)(arow1 + D_MSG + kk);
        c0 = __builtin_amdgcn_wmma_f32_16x16x4_f32(false, a0, false, b,
                                                   (short)0, c0, false, false);
        c1 = __builtin_amdgcn_wmma_f32_16x16x4_f32(false, a1, false, b,
                                                   (short)0, c1, false, false);
    }

    // Epilogue: VGPR v -> row m0 + v + 8*khalf (+16 for tile 1), col n.
    // Stores guarded for the final (partial) block only.
    #pragma unroll
    for (int vv = 0; vv < 8; ++vv) {
        int r0 = m0 + vv + khalf * 8;
        int r1 = r0 + 16;
        if (r0 < N_NODES) out[(long)r0 * D_OUT + n] = c0[vv];
        if (r1 < N_NODES) out[(long)r1 * D_OUT + n] = c1[vv];
    }
}

// ---------------------------------------------------------------------------
extern "C" void kernel_launch(void* const* d_in, const int* in_sizes, int n_in,
                              void* d_out, int out_size, void* d_ws, size_t ws_size,
                              hipStream_t stream) {
    (void)in_sizes; (void)n_in; (void)out_size; (void)ws_size;

    const float*     x_  = (const float*)d_in[0];
    const long long* ei  = (const long long*)d_in[1];   // int64 per reference
    const float*     ea  = (const float*)d_in[2];
    const float*     et  = (const float*)d_in[3];
    const float*     Wm  = (const float*)d_in[4];
    const float*     bm  = (const float*)d_in[5];
    const float*     Wr  = (const float*)d_in[6];
    const float*     br  = (const float*)d_in[7];
    float*           out = (float*)d_out;
    float*           agg = (float*)d_ws;                // 50000*96*4 = 19.2 MB

    // 0) zero agg
    const int n4 = (N_NODES * D_MSG) / 4;               // 1.2M vec4
    zero_agg_kernel<<<(n4 + 255) / 256, 256, 0, stream>>>((v4f*)agg, n4);

    // 1) scatter-sum messages into agg
    const long scat_threads = (long)N_EDGES * 24;
    const int  scat_blocks  = (int)((scat_threads + 255) / 256);
    edge_scatter_kernel<<<scat_blocks, 256, 0, stream>>>(x_, ei, ea, et, agg);

    // 2) fused WMMA GEMM + bias epilogue (two M-tiles per block)
    const int gemm_blocks = (N_NODES + ROWS_PB - 1) / ROWS_PB;  // 1563
    gemm_wmma_kernel<<<gemm_blocks, 256, 0, stream>>>(agg, x_, Wm, bm, Wr, br, out);
}